// MultinomialMaxPool2d_67783173865517
// MI455X (gfx1250) — compile-verified
//
#include <hip/hip_runtime.h>
#include <hip/hip_bf16.h>
#include <math.h>

#define KEPS 1e-8f

// clang ext_vector types: required for __builtin_nontemporal_* (HIP float4 is a struct)
typedef float f32x4 __attribute__((ext_vector_type(4)));

// --- deterministic counter-based RNG (lowbias32 avalanche hash) ---------------
__device__ __forceinline__ unsigned int mix32(unsigned int x) {
    x ^= x >> 16; x *= 0x7feb352dU;
    x ^= x >> 15; x *= 0x846ca68bU;
    x ^= x >> 16;
    return x;
}
// U in [0,1) from top 24 bits of the hash
__device__ __forceinline__ float u01(unsigned int c) {
    return (float)(mix32(c) >> 8) * (1.0f / 16777216.0f);
}

struct RegionOut {
    float s0, s1, s2, s3;   // sparse values at kernel positions (0,0),(0,1),(1,0),(1,1)
    float pooled;           // sum of non-null probs
    float widx;             // winner index as float (4 == null)
};

// Categorical sample via inverse CDF: one uniform per region.
// T = U * (e0+e1+e2+e3+en); winner = #cumsums <= T.  Same distribution as
// the reference's Gumbel-argmax over softmax probs.
__device__ __forceinline__ void process_region(float a0, float a1, float a2, float a3,
                                               unsigned int rid, RegionOut& o) {
    float m  = fmaxf(fmaxf(a0, a1), fmaxf(fmaxf(a2, a3), 0.0f));
    float e0 = __expf(a0 - m), e1 = __expf(a1 - m);
    float e2 = __expf(a2 - m), e3 = __expf(a3 - m);
    float en = __expf(-m);

    float c0 = e0;
    float c1 = c0 + e1;
    float c2 = c1 + e2;
    float c3 = c2 + e3;
    float c4 = c3 + en;                       // full partition mass
    float inv = __builtin_amdgcn_rcpf(c4 + KEPS);

    float T = u01(rid) * c4;                  // T in [0, c4)
    int g0 = (T >= c0), g1 = (T >= c1), g2 = (T >= c2), g3 = (T >= c3);
    int bi = g0 + g1 + g2 + g3;

    // branchless sequential-overwrite select: lowers to v_cmp + v_cndmask chain
    float w = e0;
    w = g0 ? e1 : w;
    w = g1 ? e2 : w;
    w = g2 ? e3 : w;
    w = g3 ? en : w;
    float p = w * inv;                        // winner probability

    o.s0 = (bi == 0) ? p : 0.0f;
    o.s1 = (bi == 1) ? p : 0.0f;
    o.s2 = (bi == 2) ? p : 0.0f;
    o.s3 = (bi == 3) ? p : 0.0f;
    o.pooled = c3 * inv;                      // sum of non-null probs
    o.widx   = (float)bi;
}

// One thread = four horizontally adjacent 2x2 regions (8 cols x 2 rows):
// all memory ops are NT B128.  H = W = 128 per reference; planes = B*C.
__global__ void __launch_bounds__(256)
mmp2d_kernel(const float* __restrict__ in,
             float* __restrict__ sparse,
             float* __restrict__ pooled,
             float* __restrict__ widx,
             int nocts) {
    const int stride = (int)(gridDim.x * blockDim.x);
    for (int t = (int)(blockIdx.x * blockDim.x + threadIdx.x); t < nocts; t += stride) {
        // decompose: 16 octs per region-row, 64 region-rows per plane
        int oc    = t & 15;          // oct (4 regions) within region-row
        int rest  = t >> 4;          // plane*64 + rh
        int rh    = rest & 63;
        int plane = rest >> 6;
        int w0    = oc << 3;         // 8 input columns
        int h0    = rh << 1;         // 2 input rows
        size_t base = (size_t)plane * 16384 + (size_t)h0 * 128 + (size_t)w0;

        // gfx1250 L2 prefetch for next grid-stride iteration's two rows
        int t2 = t + stride;
        if (t2 < nocts) {
            int oc2 = t2 & 15, rest2 = t2 >> 4;
            size_t base2 = (size_t)(rest2 >> 6) * 16384 +
                           (size_t)((rest2 & 63) << 1) * 128 + (size_t)(oc2 << 3);
            __builtin_prefetch(in + base2, 0, 3);         // row h
            __builtin_prefetch(in + base2 + 128, 0, 3);   // row h+1
        }

        // single-touch streaming loads: 4x NT B128
        f32x4 r0a = __builtin_nontemporal_load((const f32x4*)(in + base));
        f32x4 r0b = __builtin_nontemporal_load((const f32x4*)(in + base + 4));
        f32x4 r1a = __builtin_nontemporal_load((const f32x4*)(in + base + 128));
        f32x4 r1b = __builtin_nontemporal_load((const f32x4*)(in + base + 132));

        // region ids in reference order (R index = rh*64 + rw)
        unsigned int rg = (unsigned int)(plane * 4096 + rh * 64 + (oc << 2));

        RegionOut R0, R1, R2, R3;
        process_region(r0a.x, r0a.y, r1a.x, r1a.y, rg + 0u, R0);
        process_region(r0a.z, r0a.w, r1a.z, r1a.w, rg + 1u, R1);
        process_region(r0b.x, r0b.y, r1b.x, r1b.y, rg + 2u, R2);
        process_region(r0b.z, r0b.w, r1b.z, r1b.w, rg + 3u, R3);

        // sparse scatter at full resolution: 4x NT B128
        f32x4 o0a = { R0.s0, R0.s1, R1.s0, R1.s1 };
        f32x4 o0b = { R2.s0, R2.s1, R3.s0, R3.s1 };
        f32x4 o1a = { R0.s2, R0.s3, R1.s2, R1.s3 };
        f32x4 o1b = { R2.s2, R2.s3, R3.s2, R3.s3 };
        __builtin_nontemporal_store(o0a, (f32x4*)(sparse + base));
        __builtin_nontemporal_store(o0b, (f32x4*)(sparse + base + 4));
        __builtin_nontemporal_store(o1a, (f32x4*)(sparse + base + 128));
        __builtin_nontemporal_store(o1b, (f32x4*)(sparse + base + 132));

        // pooled + winner indices: one NT B128 each (16B aligned: oc*4 floats)
        size_t pbase = (size_t)plane * 4096 + (size_t)rh * 64 + (size_t)(oc << 2);
        f32x4 pp = { R0.pooled, R1.pooled, R2.pooled, R3.pooled };
        f32x4 ww = { R0.widx,   R1.widx,   R2.widx,   R3.widx   };
        __builtin_nontemporal_store(pp, (f32x4*)(pooled + pbase));
        __builtin_nontemporal_store(ww, (f32x4*)(widx + pbase));
    }
}

extern "C" void kernel_launch(void* const* d_in, const int* in_sizes, int n_in,
                              void* d_out, int out_size, void* d_ws, size_t ws_size,
                              hipStream_t stream) {
    (void)n_in; (void)out_size; (void)d_ws; (void)ws_size;
    const float* in = (const float*)d_in[0];

    const size_t total   = (size_t)in_sizes[0];   // B*C*128*128 = 50,331,648
    const size_t pooledN = total >> 2;            // 12,582,912
    float* out    = (float*)d_out;
    float* sparse = out;                          // [B,C,128,128]
    float* pooled = out + total;                  // [B,C,64,64]
    float* widx   = out + total + pooledN;        // [B,C,64,64]

    const int nocts = (int)(total >> 4);          // 4 regions (16 input elems) per thread
    dim3 block(256);                              // 8 wave32s
    dim3 grid(3072);                              // persistent grid-stride, 4 iters/thread
    hipLaunchKernelGGL(mmp2d_kernel, grid, block, 0, stream,
                       in, sparse, pooled, widx, nocts);
}